// LowPassFilter_72516227826246
// MI455X (gfx1250) — compile-verified
//
#include <hip/hip_runtime.h>
#include <cstddef>

typedef __attribute__((ext_vector_type(2))) float v2f;
typedef __attribute__((ext_vector_type(8))) float v8f;

// Only probe for the builtin during the DEVICE compilation pass -- the host
// pass has no amdgcn builtins and would always trip this check.
#if defined(__HIP_DEVICE_COMPILE__) && !__has_builtin(__builtin_amdgcn_wmma_f32_16x16x4_f32)
#error "__builtin_amdgcn_wmma_f32_16x16x4_f32 not available on gfx1250 device pass"
#endif

static __device__ __forceinline__ v8f wmma4(v2f a, v2f b, v8f c) {
  // D = A(16x4,f32) * B(4x16,f32) + C(16x16,f32)
  return __builtin_amdgcn_wmma_f32_16x16x4_f32(false, a, false, b, (short)0, c,
                                               false, false);
}

// ---------------- problem constants ----------------
constexpr int NROWS = 256;
constexpr int TLEN  = 131072;
constexpr int PAD   = 15;                   // scipy filtfilt padlen
constexpr int ELEN  = TLEN + 2 * PAD;       // 131102 extended length
constexpr int ES    = 131104;               // ELEN rounded up to 16 (yf stride)
constexpr int NBLK  = ES / 16;              // 8194 blocks of 16 samples
constexpr int CB    = 128;                  // blocks per chunk (2048 samples) -> 1040 waves/pass
constexpr int WB    = 16;                   // warm-up blocks (256 samples; |p|max^256 ~ 6e-14)
constexpr int NCHUNK = (NBLK + CB - 1) / CB; // 65
constexpr int NRG   = NROWS / 16;           // 16 row-groups (16 rows per wave)

// ---------------- workspace layout (floats) ----------------
constexpr int WS_TB = 0;      // 4 chunks x 64  : Toeplitz(h) B-operands
constexpr int WS_WB = 256;    // 64             : state->output B-operand
constexpr int WS_UB = 320;    // 4 chunks x 64  : input->state  B-operands
constexpr int WS_PB = 576;    // 64             : A^16 B-operand
constexpr int WS_ZI = 640;    // 4              : lfilter_zi
constexpr int WS_YF = 1024;   // NROWS * ES     : forward-pass intermediate (L2-resident)

// ================= setup: derive block state-space matrices =================
// DF2T:  z' = A z + Bv x,  y = z0 + b0 x
// A = [[-a1,1,0,0],[-a2,0,1,0],[-a3,0,0,1],[-a4,0,0,0]], Bv = b[1:] - a[1:]*b0
// 16-sample block:  Y = X*T^t + Z*W^t ,  Z' = X*U^t + Z*P^t
//   T[t][j] = h[t-j] (lower tri), h[0]=b0, h[m]=C A^{m-1} Bv
//   W[t]    = C A^t ,  U[:,j] = A^{15-j} Bv ,  P = A^16
// B-operand lane image (mirrors documented A layout):
//   element v, lane L holds B[k][n] with k = v + 2*(L>=16), n = L%16
__global__ void setup_kernel(const float* __restrict__ bc,
                             const float* __restrict__ ac,
                             float* __restrict__ ws) {
  if (threadIdx.x != 0 || blockIdx.x != 0) return;
  double a0 = (double)ac[0];
  double an[5], bn[5];
  for (int i = 0; i < 5; ++i) { an[i] = (double)ac[i] / a0; bn[i] = (double)bc[i] / a0; }
  const double b0 = bn[0];
  double Bv[4];
  for (int i = 0; i < 4; ++i) Bv[i] = bn[i + 1] - an[i + 1] * b0;
  double A[4][4] = {{-an[1], 1, 0, 0},
                    {-an[2], 0, 1, 0},
                    {-an[3], 0, 0, 1},
                    {-an[4], 0, 0, 0}};
  // Crow[k] = C * A^k
  double Crow[16][4];
  Crow[0][0] = 1; Crow[0][1] = 0; Crow[0][2] = 0; Crow[0][3] = 0;
  for (int k = 1; k < 16; ++k)
    for (int j = 0; j < 4; ++j) {
      double s = 0;
      for (int i = 0; i < 4; ++i) s += Crow[k - 1][i] * A[i][j];
      Crow[k][j] = s;
    }
  // impulse response
  double h[16];
  h[0] = b0;
  for (int m = 1; m < 16; ++m) {
    double s = 0;
    for (int i = 0; i < 4; ++i) s += Crow[m - 1][i] * Bv[i];
    h[m] = s;
  }
  // u[p] = A^p Bv
  double u[16][4];
  for (int i = 0; i < 4; ++i) u[0][i] = Bv[i];
  for (int p = 1; p < 16; ++p)
    for (int i = 0; i < 4; ++i) {
      double s = 0;
      for (int j = 0; j < 4; ++j) s += A[i][j] * u[p - 1][j];
      u[p][i] = s;
    }
  // P = A^16
  double P[4][4], Tm[4][4];
  for (int i = 0; i < 4; ++i)
    for (int j = 0; j < 4; ++j) P[i][j] = (i == j) ? 1.0 : 0.0;
  for (int s = 0; s < 16; ++s) {
    for (int i = 0; i < 4; ++i)
      for (int j = 0; j < 4; ++j) {
        double t = 0;
        for (int q = 0; q < 4; ++q) t += A[i][q] * P[q][j];
        Tm[i][j] = t;
      }
    for (int i = 0; i < 4; ++i)
      for (int j = 0; j < 4; ++j) P[i][j] = Tm[i][j];
  }
  // zi: (I - comp^T) zi = Bv  (4x4 Gaussian elimination w/ pivoting)
  double M[4][5];
  for (int i = 0; i < 4; ++i) {
    for (int j = 0; j < 4; ++j)
      M[i][j] = ((i == j) ? 1.0 : 0.0) + ((j == 0) ? an[i + 1] : 0.0) -
                ((j == i + 1) ? 1.0 : 0.0);
    M[i][4] = Bv[i];
  }
  for (int c = 0; c < 4; ++c) {
    int piv = c;
    for (int r = c + 1; r < 4; ++r)
      if ((M[r][c] < 0 ? -M[r][c] : M[r][c]) > (M[piv][c] < 0 ? -M[piv][c] : M[piv][c]))
        piv = r;
    if (piv != c)
      for (int j = 0; j < 5; ++j) { double t = M[c][j]; M[c][j] = M[piv][j]; M[piv][j] = t; }
    double d = M[c][c];
    for (int j = 0; j < 5; ++j) M[c][j] /= d;
    for (int r = 0; r < 4; ++r) {
      if (r == c) continue;
      double f = M[r][c];
      for (int j = 0; j < 5; ++j) M[r][j] -= f * M[c][j];
    }
  }
  // emit B-operand images
  for (int c = 0; c < 4; ++c)
    for (int v = 0; v < 2; ++v)
      for (int L = 0; L < 32; ++L) {
        int k = v + ((L >= 16) ? 2 : 0);
        int n = L & 15;
        int kk = 4 * c + k;
        ws[WS_TB + c * 64 + v * 32 + L] = (float)((n >= kk) ? h[n - kk] : 0.0);
        ws[WS_UB + c * 64 + v * 32 + L] = (float)((n < 4) ? u[15 - kk][n] : 0.0);
      }
  for (int v = 0; v < 2; ++v)
    for (int L = 0; L < 32; ++L) {
      int k = v + ((L >= 16) ? 2 : 0);
      int n = L & 15;
      ws[WS_WB + v * 32 + L] = (float)Crow[n][k];
      ws[WS_PB + v * 32 + L] = (float)((n < 4) ? P[n][k] : 0.0);
    }
  for (int i = 0; i < 4; ++i) ws[WS_ZI + i] = (float)M[i][4];
}

// odd-reflect extension of a row, ext index e in [0, ES)
static __device__ __forceinline__ float load_ext_fwd(const float* __restrict__ xr, int e) {
  if (e < PAD) return 2.0f * xr[0] - xr[PAD - e];
  if (e < PAD + TLEN) return xr[e - PAD];
  int j = e - (PAD + TLEN);
  if (j > PAD - 1) j = PAD - 1;  // clamp pad-to-16 overrun (values unused)
  return 2.0f * xr[TLEN - 1] - xr[TLEN - 2 - j];
}

static __device__ __forceinline__ float load_rev(const float* __restrict__ yr, int e) {
  int idx = ELEN - 1 - e;
  if (idx < 0) idx = 0;  // pad-to-16 overrun (values unused)
  return yr[idx];
}

// ===================== forward IIR pass (block state-space) =====================
__global__ __launch_bounds__(32) void fwd_kernel(const float* __restrict__ x,
                                                 float* __restrict__ ws) {
  float* __restrict__ yf = ws + WS_YF;
  const int task = blockIdx.x;
  const int rg = task / NCHUNK, ch = task % NCHUNK;
  const int r0 = rg * 16;
  const int L = threadIdx.x;
  const int m = L & 15;        // A-operand row (two lanes per row)
  const int hiA = L >> 4;      // lane half
  const int k0 = 2 * hiA;      // A-operand K base: v0->k0, v1->k0+1
  const float* __restrict__ xrow = x + (size_t)(r0 + m) * TLEN;

  v2f TBv[4], UBv[4], WBv, PBv;
#pragma unroll
  for (int c = 0; c < 4; ++c) {
    TBv[c] = (v2f){ws[WS_TB + c * 64 + L], ws[WS_TB + c * 64 + 32 + L]};
    UBv[c] = (v2f){ws[WS_UB + c * 64 + L], ws[WS_UB + c * 64 + 32 + L]};
  }
  WBv = (v2f){ws[WS_WB + L], ws[WS_WB + 32 + L]};
  PBv = (v2f){ws[WS_PB + L], ws[WS_PB + 32 + L]};
  const float zi0 = ws[WS_ZI + k0], zi1 = ws[WS_ZI + k0 + 1];

  const int bOut = ch * CB;
  const int bW = (bOut - WB > 0) ? (bOut - WB) : 0;
  const int bEnd = ((bOut + CB) < NBLK) ? (bOut + CB) : NBLK;

  float v0 = load_ext_fwd(xrow, bW * 16);
  v2f z = (v2f){zi0 * v0, zi1 * v0};  // exact for ch==0; warm-up decays for interior chunks

  // D-layout(Z') -> A-layout(Z) shuffle: Z[m][k] lives at lane k+16*(m>=8), elem m%8
  const int sl0 = k0 + ((m & 8) ? 16 : 0);
  const int sl1 = sl0 + 1;
  const int vsel = L & 7;

  for (int bb = bW; bb < bEnd; ++bb) {
    const int e0 = bb * 16;
    v2f xa[4];
    if (e0 >= PAD && (e0 + 16) <= (PAD + TLEN)) {  // interior fast path
      const float* p = xrow + (e0 - PAD) + k0;
#pragma unroll
      for (int c = 0; c < 4; ++c) xa[c] = (v2f){p[4 * c], p[4 * c + 1]};
      __builtin_prefetch(p + 64, 0, 1);  // 4 blocks ahead -> global_prefetch
    } else {
#pragma unroll
      for (int c = 0; c < 4; ++c) {
        int o = e0 + 4 * c + k0;
        xa[c] = (v2f){load_ext_fwd(xrow, o), load_ext_fwd(xrow, o + 1)};
      }
    }
    const v8f zero = (v8f){0, 0, 0, 0, 0, 0, 0, 0};
    v8f dZ = wmma4(z, PBv, zero);          // Z' = Z*P^t + X*U^t  (cols 0..3 valid)
#pragma unroll
    for (int c = 0; c < 4; ++c) dZ = wmma4(xa[c], UBv[c], dZ);
    if (bb >= bOut) {                      // warm-up blocks skip the output half
      v8f dY = wmma4(z, WBv, zero);        // Y = Z*W^t + X*T^t
#pragma unroll
      for (int c = 0; c < 4; ++c) dY = wmma4(xa[c], TBv[c], dY);
#pragma unroll
      for (int v = 0; v < 8; ++v)
        yf[(size_t)(r0 + v + 8 * hiA) * ES + e0 + m] = dY[v];
    }
    float nz0 = 0.f, nz1 = 0.f;
#pragma unroll
    for (int v = 0; v < 8; ++v) {
      float g0 = __shfl(dZ[v], sl0, 32);
      float g1 = __shfl(dZ[v], sl1, 32);
      nz0 = (vsel == v) ? g0 : nz0;
      nz1 = (vsel == v) ? g1 : nz1;
    }
    z = (v2f){nz0, nz1};
  }
}

// ===================== backward IIR pass over reversed yf =====================
__global__ __launch_bounds__(32) void bwd_kernel(const float* __restrict__ ws,
                                                 float* __restrict__ out) {
  const float* __restrict__ yf = ws + WS_YF;
  const int task = blockIdx.x;
  const int rg = task / NCHUNK, ch = task % NCHUNK;
  const int r0 = rg * 16;
  const int L = threadIdx.x;
  const int m = L & 15;
  const int hiA = L >> 4;
  const int k0 = 2 * hiA;
  const float* __restrict__ yrow = yf + (size_t)(r0 + m) * ES;

  v2f TBv[4], UBv[4], WBv, PBv;
#pragma unroll
  for (int c = 0; c < 4; ++c) {
    TBv[c] = (v2f){ws[WS_TB + c * 64 + L], ws[WS_TB + c * 64 + 32 + L]};
    UBv[c] = (v2f){ws[WS_UB + c * 64 + L], ws[WS_UB + c * 64 + 32 + L]};
  }
  WBv = (v2f){ws[WS_WB + L], ws[WS_WB + 32 + L]};
  PBv = (v2f){ws[WS_PB + L], ws[WS_PB + 32 + L]};
  const float zi0 = ws[WS_ZI + k0], zi1 = ws[WS_ZI + k0 + 1];

  const int bOut = ch * CB;
  const int bW = (bOut - WB > 0) ? (bOut - WB) : 0;
  const int bEnd = ((bOut + CB) < NBLK) ? (bOut + CB) : NBLK;

  float v0 = load_rev(yrow, bW * 16);   // ch==0: zi * yf[E-1], exact
  v2f z = (v2f){zi0 * v0, zi1 * v0};
  const int sl0 = k0 + ((m & 8) ? 16 : 0);
  const int sl1 = sl0 + 1;
  const int vsel = L & 7;

  for (int bb = bW; bb < bEnd; ++bb) {
    const int e0 = bb * 16;
    v2f xa[4];
    if (e0 + 16 <= ELEN) {
      const float* p = yrow + (ELEN - 1 - e0 - k0);
#pragma unroll
      for (int c = 0; c < 4; ++c) xa[c] = (v2f){p[-4 * c], p[-4 * c - 1]};
      __builtin_prefetch(p - 64, 0, 1);  // 4 blocks ahead (descending stream)
    } else {
#pragma unroll
      for (int c = 0; c < 4; ++c) {
        int o = e0 + 4 * c + k0;
        xa[c] = (v2f){load_rev(yrow, o), load_rev(yrow, o + 1)};
      }
    }
    const v8f zero = (v8f){0, 0, 0, 0, 0, 0, 0, 0};
    v8f dZ = wmma4(z, PBv, zero);
#pragma unroll
    for (int c = 0; c < 4; ++c) dZ = wmma4(xa[c], UBv[c], dZ);
    if (bb >= bOut) {
      v8f dY = wmma4(z, WBv, zero);
#pragma unroll
      for (int c = 0; c < 4; ++c) dY = wmma4(xa[c], TBv[c], dY);
      const int e = e0 + m;
      if (e >= PAD && e < PAD + TLEN) {   // out[t] = yb[E-16-t] => col = E-16-e
        const int col = ELEN - 16 - e;
#pragma unroll
        for (int v = 0; v < 8; ++v)
          out[(size_t)(r0 + v + 8 * hiA) * TLEN + col] = dY[v];
      }
    }
    float nz0 = 0.f, nz1 = 0.f;
#pragma unroll
    for (int v = 0; v < 8; ++v) {
      float g0 = __shfl(dZ[v], sl0, 32);
      float g1 = __shfl(dZ[v], sl1, 32);
      nz0 = (vsel == v) ? g0 : nz0;
      nz1 = (vsel == v) ? g1 : nz1;
    }
    z = (v2f){nz0, nz1};
  }
}

extern "C" void kernel_launch(void* const* d_in, const int* in_sizes, int n_in,
                              void* d_out, int out_size, void* d_ws, size_t ws_size,
                              hipStream_t stream) {
  (void)in_sizes; (void)n_in; (void)out_size; (void)ws_size;
  const float* x = (const float*)d_in[0];
  const float* b = (const float*)d_in[1];
  const float* a = (const float*)d_in[2];
  float* out = (float*)d_out;
  float* ws = (float*)d_ws;  // needs (1024 + 256*ES)*4 B ~= 128.06 MiB (L2-resident)

  setup_kernel<<<1, 1, 0, stream>>>(b, a, ws);
  fwd_kernel<<<NRG * NCHUNK, 32, 0, stream>>>(x, ws);
  bwd_kernel<<<NRG * NCHUNK, 32, 0, stream>>>(ws, out);
}